// MyNet_16338055594085
// MI455X (gfx1250) — compile-verified
//
#include <hip/hip_runtime.h>
#include <hip/hip_bf16.h>

#define N_NODES 20000
#define N_EDGES 640000

typedef __attribute__((ext_vector_type(2))) float v2f;
typedef __attribute__((ext_vector_type(8))) float v8f;

__device__ __forceinline__ v8f wmma_f32(v2f a, v2f b, v8f c) {
  // V_WMMA_F32_16X16X4_F32 : D(16x16 f32) = A(16x4 f32) * B(4x16 f32) + C
  return __builtin_amdgcn_wmma_f32_16x16x4_f32(
      /*neg_a=*/false, a, /*neg_b=*/false, b,
      /*c_mod=*/(short)0, c, /*reuse_a=*/false, /*reuse_b=*/false);
}

__device__ __forceinline__ float sigm(float x) {
  return 1.0f / (1.0f + __expf(-x));
}

// ---------------------------------------------------------------------------
// Fused 2-layer LSTM over 24 steps. One block = 16 sequences (nodes).
// 512 threads = 16 waves; wave w owns gate columns [16w, 16w+16).
// h/c state resident in LDS across all timesteps.
// ---------------------------------------------------------------------------
__global__ __launch_bounds__(512) void lstm_fused_kernel(
    const float* __restrict__ seq,                 // [N][24][8]
    const float* __restrict__ Wih0,                // [256][8]
    const float* __restrict__ Whh0,                // [256][64]
    const float* __restrict__ bih0,                // [256]
    const float* __restrict__ bhh0,                // [256]
    const float* __restrict__ Wih1,                // [256][64]
    const float* __restrict__ Whh1,                // [256][64]
    const float* __restrict__ bih1,                // [256]
    const float* __restrict__ bhh1,                // [256]
    float* __restrict__ xtemp)                     // [N] mean of final h1
{
  __shared__ float sA[16][80];    // layer0 A = [x_t (0:8) | h0 (8:72)]
  __shared__ float sB[16][136];   // layer1 A = [h0_cur (0:64) | h1 (64:128)]
  __shared__ float sG[16][264];   // gate buffer 16x256
  __shared__ float sC0[16][64];
  __shared__ float sC1[16][64];

  const int tid  = threadIdx.x;
  const int lane = tid & 31;
  const int wave = tid >> 5;           // 0..15
  const int nodeBase = blockIdx.x * 16;

  // zero initial h/c state
  for (int i = tid; i < 16 * 64; i += 512) {
    int r = i >> 6, c = i & 63;
    sA[r][8 + c] = 0.0f;
    sB[r][64 + c] = 0.0f;
    sC0[r][c] = 0.0f;
    sC1[r][c] = 0.0f;
  }
  __syncthreads();

  const int mlane = lane & 15;         // M index (row of tile) / N index
  const int kl2   = (lane >> 4) * 2;   // K sub-offset of this half-wave
  const int r0    = (lane >> 4) * 8;   // C/D: lanes>=16 hold M=8..15
  const int n     = wave * 16 + mlane; // this lane's gate column
  const float b0  = bih0[n] + bhh0[n];
  const float b1  = bih1[n] + bhh1[n];

  for (int t = 0; t < 24; ++t) {
    // stage x_t for the 16 rows
    if (tid < 128) {
      int r = tid >> 3, c = tid & 7;
      sA[r][c] = seq[((long)(nodeBase + r) * 24 + t) * 8 + c];
    }
    __syncthreads();

    // ---- layer 0 gates: g = [x|h0] @ [Wih0|Whh0]^T  (K = 72) ----
    {
      v8f acc = {};
      const float* wih = Wih0 + n * 8  + kl2;
      const float* whh = Whh0 + n * 64 + kl2 - 8;
#pragma unroll
      for (int k = 0; k < 72; k += 4) {
        v2f a = *(const v2f*)&sA[mlane][k + kl2];
        v2f b = (k < 8) ? *(const v2f*)(wih + k) : *(const v2f*)(whh + k);
        acc = wmma_f32(a, b, acc);
      }
#pragma unroll
      for (int r = 0; r < 8; ++r) sG[r0 + r][n] = acc[r] + b0;
    }
    __syncthreads();

    // ---- layer 0 activations; each thread handles 2 of 16x64 cells ----
#pragma unroll
    for (int i = 0; i < 2; ++i) {
      int e = tid * 2 + i;
      int r = e >> 6, c = e & 63;
      float gi = sG[r][c], gf = sG[r][64 + c];
      float gg = sG[r][128 + c], go = sG[r][192 + c];
      float cc = sC0[r][c];
      cc = sigm(gf) * cc + sigm(gi) * tanhf(gg);
      float hv = sigm(go) * tanhf(cc);
      sC0[r][c] = cc;
      sA[r][8 + c] = hv;   // h0 for next step's layer0
      sB[r][c]     = hv;   // h0_cur for layer1
    }
    __syncthreads();

    // ---- layer 1 gates: g = [h0|h1] @ [Wih1|Whh1]^T  (K = 128) ----
    {
      v8f acc = {};
      const float* wih = Wih1 + n * 64 + kl2;
      const float* whh = Whh1 + n * 64 + kl2 - 64;
#pragma unroll
      for (int k = 0; k < 128; k += 4) {
        v2f a = *(const v2f*)&sB[mlane][k + kl2];
        v2f b = (k < 64) ? *(const v2f*)(wih + k) : *(const v2f*)(whh + k);
        acc = wmma_f32(a, b, acc);
      }
#pragma unroll
      for (int r = 0; r < 8; ++r) sG[r0 + r][n] = acc[r] + b1;
    }
    __syncthreads();

    // ---- layer 1 activations ----
#pragma unroll
    for (int i = 0; i < 2; ++i) {
      int e = tid * 2 + i;
      int r = e >> 6, c = e & 63;
      float gi = sG[r][c], gf = sG[r][64 + c];
      float gg = sG[r][128 + c], go = sG[r][192 + c];
      float cc = sC1[r][c];
      cc = sigm(gf) * cc + sigm(gi) * tanhf(gg);
      float hv = sigm(go) * tanhf(cc);
      sC1[r][c] = cc;
      sB[r][64 + c] = hv;
    }
    __syncthreads();
  }

  // x_temporal = mean over 64 hidden dims of final h1
  if (tid < 16) {
    float s = 0.0f;
#pragma unroll 8
    for (int c = 0; c < 64; ++c) s += sB[tid][64 + c];
    xtemp[nodeBase + tid] = s * (1.0f / 64.0f);
  }
}

// ---------------------------------------------------------------------------
// Generic WMMA-f32 GEMM: out[M x Nout] = relu?(A[M x Kin] @ W^T + bias)
// W is [Nout x Kin] row-major. One block = one 16-row tile; 8 waves split
// the Nout/16 column tiles. A and B fragments are aligned float2 loads.
// Requires Kin % 4 == 0, lda/ldw even.
// ---------------------------------------------------------------------------
__global__ __launch_bounds__(256) void gemm_wmma_kernel(
    const float* __restrict__ A, int lda,
    const float* __restrict__ W, int ldw,
    const float* __restrict__ bias,
    float* __restrict__ out, int ldo,
    int Nout, int Kin, int do_relu)
{
  const int lane = threadIdx.x & 31;
  const int wave = threadIdx.x >> 5;
  const int nwaves = blockDim.x >> 5;
  const int mbase = blockIdx.x * 16;
  const int mlane = lane & 15;
  const int kl2 = (lane >> 4) * 2;
  const int r0  = (lane >> 4) * 8;
  const int ntiles = Nout >> 4;

  const float* arow = A + (long)(mbase + mlane) * lda + kl2;

  for (int nt = wave; nt < ntiles; nt += nwaves) {
    const int nn = nt * 16 + mlane;
    const float* wrow = W + (long)nn * ldw + kl2;
    v8f acc = {};
    for (int k = 0; k < Kin; k += 4) {
      v2f a = *(const v2f*)(arow + k);
      v2f b = *(const v2f*)(wrow + k);
      acc = wmma_f32(a, b, acc);
    }
    const float bv = bias ? bias[nn] : 0.0f;
#pragma unroll
    for (int r = 0; r < 8; ++r) {
      float v = acc[r] + bv;
      if (do_relu) v = fmaxf(v, 0.0f);
      out[(long)(mbase + r0 + r) * ldo + nn] = v;
    }
  }
}

// ---------------------------------------------------------------------------
// GCN support kernels
// ---------------------------------------------------------------------------
__global__ void deg_init_kernel(float* __restrict__ deg, int n) {
  int i = blockIdx.x * blockDim.x + threadIdx.x;
  if (i < n) deg[i] = 1.0f;  // self loop
}

__global__ void deg_acc_kernel(const long long* __restrict__ dst,
                               float* __restrict__ deg, int e) {
  int i = blockIdx.x * blockDim.x + threadIdx.x;
  if (i < e) atomicAdd(&deg[dst[i]], 1.0f);
}

__global__ void deg_rsqrt_kernel(float* __restrict__ deg, int n) {
  int i = blockIdx.x * blockDim.x + threadIdx.x;
  if (i < n) {
    float d = deg[i];
    deg[i] = (d > 0.0f) ? rsqrtf(d) : 0.0f;
  }
}

// out[n][c] = h[n][c] * dinv[n]^2   (self-loop message; also zero-inits out)
__global__ void self_init_kernel(const float* __restrict__ h,
                                 const float* __restrict__ dinv,
                                 float* __restrict__ out, int total) {
  int i = blockIdx.x * blockDim.x + threadIdx.x;
  if (i < total) {
    int node = i >> 7;
    float dv = dinv[node];
    out[i] = h[i] * dv * dv;
  }
}

// edge scatter: out[dst] += h[src] * dinv[src]*dinv[dst]; 4 channels/thread
__global__ void scatter_kernel(const float* __restrict__ h,
                               const float* __restrict__ dinv,
                               const long long* __restrict__ src,
                               const long long* __restrict__ dst,
                               float* __restrict__ out, int e) {
  int idx = blockIdx.x * blockDim.x + threadIdx.x;
  int ei = idx >> 5;
  if (ei >= e) return;
  int c = (idx & 31) * 4;
  long long s = src[ei], d = dst[ei];
  float coef = dinv[s] * dinv[d];
  const float4 hv = *(const float4*)&h[s * 128 + c];
  float* o = &out[d * 128 + c];
  atomicAdd(o + 0, hv.x * coef);
  atomicAdd(o + 1, hv.y * coef);
  atomicAdd(o + 2, hv.z * coef);
  atomicAdd(o + 3, hv.w * coef);
}

__global__ void bias_relu_kernel(float* __restrict__ s,
                                 const float* __restrict__ b, int total) {
  int i = blockIdx.x * blockDim.x + threadIdx.x;
  if (i < total) s[i] = fmaxf(s[i] + b[i & 127], 0.0f);
}

// mcat[n][0:132] = relu([g2(128) | xt(1) | external(2) | 0-pad(1)])
__global__ void concat_kernel(const float* __restrict__ g2,
                              const float* __restrict__ xt,
                              const float* __restrict__ x,
                              float* __restrict__ mcat, int total) {
  int i = blockIdx.x * blockDim.x + threadIdx.x;
  if (i >= total) return;
  int node = i / 132;
  int c = i - node * 132;
  float v;
  if (c < 128)       v = g2[node * 128 + c];
  else if (c == 128) v = xt[node];
  else if (c < 131)  v = x[node * 66 + (c - 129)];
  else               v = 0.0f;
  mcat[i] = fmaxf(v, 0.0f);
}

// zero-pad lw1 [64][131] -> lw1p [64][132]
__global__ void padw_kernel(const float* __restrict__ lw1,
                            float* __restrict__ lw1p) {
  int i = blockIdx.x * blockDim.x + threadIdx.x;
  if (i >= 64 * 132) return;
  int nrow = i / 132;
  int c = i - nrow * 132;
  lw1p[i] = (c < 131) ? lw1[nrow * 131 + c] : 0.0f;
}

// out[n] = dot(m2[n], lw3) + lb3
__global__ void final_kernel(const float* __restrict__ m2,
                             const float* __restrict__ lw3,
                             const float* __restrict__ lb3,
                             float* __restrict__ out, int n) {
  int i = blockIdx.x * blockDim.x + threadIdx.x;
  if (i >= n) return;
  float s = 0.0f;
#pragma unroll 8
  for (int k = 0; k < 64; ++k) s += m2[i * 64 + k] * lw3[k];
  out[i] = s + lb3[0];
}

// ---------------------------------------------------------------------------
extern "C" void kernel_launch(void* const* d_in, const int* in_sizes, int n_in,
                              void* d_out, int out_size, void* d_ws, size_t ws_size,
                              hipStream_t stream) {
  const float*     x    = (const float*)d_in[0];      // [20000][66]
  const long long* ei   = (const long long*)d_in[1];  // [2][E] int64
  const float*     seq  = (const float*)d_in[2];      // [20000][24][8]
  const float*     gW1  = (const float*)d_in[3];      // [128][64]
  const float*     gb1  = (const float*)d_in[4];
  const float*     gW2  = (const float*)d_in[5];      // [128][128]
  const float*     gb2  = (const float*)d_in[6];
  const float*     Wih0 = (const float*)d_in[7];
  const float*     Whh0 = (const float*)d_in[8];
  const float*     bih0 = (const float*)d_in[9];
  const float*     bhh0 = (const float*)d_in[10];
  const float*     Wih1 = (const float*)d_in[11];
  const float*     Whh1 = (const float*)d_in[12];
  const float*     bih1 = (const float*)d_in[13];
  const float*     bhh1 = (const float*)d_in[14];
  const float*     lw1  = (const float*)d_in[15];     // [64][131]
  const float*     lb1  = (const float*)d_in[16];
  const float*     lw2  = (const float*)d_in[17];     // [64][64]
  const float*     lb2  = (const float*)d_in[18];
  const float*     lw3  = (const float*)d_in[19];     // [1][64]
  const float*     lb3  = (const float*)d_in[20];

  const long long* esrc = ei;
  const long long* edst = ei + N_EDGES;

  float* ws   = (float*)d_ws;
  float* xt   = ws;                                // 20000
  float* dinv = ws + 20000;                        // 20000
  float* bufH = ws + 40000;                        // 20000*128
  float* bufS = bufH + (size_t)N_NODES * 128;      // 20000*128
  float* mcat = bufS + (size_t)N_NODES * 128;      // 20000*132
  float* lw1p = mcat + (size_t)N_NODES * 132;      // 64*132
  float* m1   = bufH;                              // reuse: 20000*64
  float* m2   = bufH + (size_t)N_NODES * 64;       // reuse: 20000*64
  float* out  = (float*)d_out;                     // [20000]

  const int MT = N_NODES / 16;                     // 1250 row tiles

  // 1) fused 2-layer LSTM -> x_temporal
  lstm_fused_kernel<<<MT, 512, 0, stream>>>(seq, Wih0, Whh0, bih0, bhh0,
                                            Wih1, Whh1, bih1, bhh1, xt);

  // 2) degree -> dinv
  deg_init_kernel<<<(N_NODES + 255) / 256, 256, 0, stream>>>(dinv, N_NODES);
  deg_acc_kernel<<<(N_EDGES + 255) / 256, 256, 0, stream>>>(edst, dinv, N_EDGES);
  deg_rsqrt_kernel<<<(N_NODES + 255) / 256, 256, 0, stream>>>(dinv, N_NODES);

  // 3) GCN layer 1: h = feat @ gW1^T ; scatter ; relu(+b)
  gemm_wmma_kernel<<<MT, 256, 0, stream>>>(x + 2, 66, gW1, 64, nullptr,
                                           bufH, 128, 128, 64, 0);
  self_init_kernel<<<(N_NODES * 128 + 255) / 256, 256, 0, stream>>>(
      bufH, dinv, bufS, N_NODES * 128);
  scatter_kernel<<<(N_EDGES * 32 + 255) / 256, 256, 0, stream>>>(
      bufH, dinv, esrc, edst, bufS, N_EDGES);
  bias_relu_kernel<<<(N_NODES * 128 + 255) / 256, 256, 0, stream>>>(
      bufS, gb1, N_NODES * 128);

  // 4) GCN layer 2
  gemm_wmma_kernel<<<MT, 256, 0, stream>>>(bufS, 128, gW2, 128, nullptr,
                                           bufH, 128, 128, 128, 0);
  self_init_kernel<<<(N_NODES * 128 + 255) / 256, 256, 0, stream>>>(
      bufH, dinv, bufS, N_NODES * 128);
  scatter_kernel<<<(N_EDGES * 32 + 255) / 256, 256, 0, stream>>>(
      bufH, dinv, esrc, edst, bufS, N_EDGES);
  bias_relu_kernel<<<(N_NODES * 128 + 255) / 256, 256, 0, stream>>>(
      bufS, gb2, N_NODES * 128);

  // 5) concat + relu (K padded 131 -> 132), pad lw1
  concat_kernel<<<(N_NODES * 132 + 255) / 256, 256, 0, stream>>>(
      bufS, xt, x, mcat, N_NODES * 132);
  padw_kernel<<<(64 * 132 + 255) / 256, 256, 0, stream>>>(lw1, lw1p);

  // 6) MLP
  gemm_wmma_kernel<<<MT, 256, 0, stream>>>(mcat, 132, lw1p, 132, lb1,
                                           m1, 64, 64, 132, 1);
  gemm_wmma_kernel<<<MT, 256, 0, stream>>>(m1, 64, lw2, 64, lb2,
                                           m2, 64, 64, 64, 1);
  final_kernel<<<(N_NODES + 255) / 256, 256, 0, stream>>>(m2, lw3, lb3,
                                                          out, N_NODES);
}